// LabelPropagation_24867860643981
// MI455X (gfx1250) — compile-verified
//
#include <hip/hip_runtime.h>

#define N_NODES   16384
#define N_CLASSES 64
#define N_EDGES   524288
#define KSTEPS    10
#define ALPHA     0.9f

typedef __attribute__((ext_vector_type(2))) float v2f;
typedef __attribute__((ext_vector_type(8))) float v8f;

// ---------------------------------------------------------------- CSR build

__global__ void zero_u32(int* __restrict__ p, int n) {
    int i = blockIdx.x * blockDim.x + threadIdx.x;
    if (i < n) p[i] = 0;
}

// cnt[dst] += 1 (integer atomics: deterministic)
__global__ void count_kernel(const int* __restrict__ dst, int* __restrict__ cnt) {
    int e = blockIdx.x * blockDim.x + threadIdx.x;
    if (e < N_EDGES) atomicAdd(&cnt[dst[e]], 1);
}

__global__ void dinv_kernel(const int* __restrict__ cnt, float* __restrict__ dinv) {
    int i = blockIdx.x * blockDim.x + threadIdx.x;
    if (i < N_NODES) dinv[i] = rsqrtf(fmaxf((float)cnt[i], 1.0f));
}

// Single-block exclusive scan of cnt[N_NODES] -> row_start[N_NODES+1], cursor copy.
// 256 threads, each owns 64 consecutive nodes.
__global__ void scan_kernel(const int* __restrict__ cnt, int* __restrict__ row_start,
                            int* __restrict__ cursor) {
    __shared__ int sh[256];
    const int t = threadIdx.x;
    int local = 0;
#pragma unroll
    for (int j = 0; j < 64; ++j) local += cnt[t * 64 + j];
    sh[t] = local;
    __syncthreads();
    // Hillis-Steele inclusive scan over 256 partials
    for (int off = 1; off < 256; off <<= 1) {
        int v = (t >= off) ? sh[t - off] : 0;
        __syncthreads();
        sh[t] += v;
        __syncthreads();
    }
    int run = sh[t] - local;   // exclusive base for this thread's span
    for (int j = 0; j < 64; ++j) {
        int i = t * 64 + j;
        row_start[i] = run;
        cursor[i]    = run;
        run += cnt[i];
    }
    if (t == 255) row_start[N_NODES] = run;   // == N_EDGES
}

// place edge ids into CSR slots (arrival order; sorted deterministically after)
__global__ void fill_kernel(const int* __restrict__ dst, int* __restrict__ cursor,
                            int* __restrict__ csr_eid) {
    int e = blockIdx.x * blockDim.x + threadIdx.x;
    if (e < N_EDGES) {
        int pos = atomicAdd(&cursor[dst[e]], 1);
        csr_eid[pos] = e;
    }
}

// Per-node: insertion-sort edge ids (avg degree 32) so summation order is a
// pure function of the input, then materialize src index + DAD weight.
__global__ void sort_mat_kernel(const int* __restrict__ row_start, int* __restrict__ csr_eid,
                                const int* __restrict__ src, const float* __restrict__ dinv,
                                int* __restrict__ csr_src, float* __restrict__ csr_w) {
    int d = blockIdx.x * blockDim.x + threadIdx.x;
    if (d >= N_NODES) return;
    int lo = row_start[d], hi = row_start[d + 1];
    for (int i = lo + 1; i < hi; ++i) {
        int v = csr_eid[i];
        int j = i - 1;
        while (j >= lo && csr_eid[j] > v) { csr_eid[j + 1] = csr_eid[j]; --j; }
        csr_eid[j + 1] = v;
    }
    float dd = dinv[d];
    for (int i = lo; i < hi; ++i) {
        int s = src[csr_eid[i]];
        csr_src[i] = s;
        csr_w[i]   = dinv[s] * dd;
    }
}

// ------------------------------------------------------- fused propagation
// out[d,c] = 0.1*in[d,c] + 0.9 * sum_{e in incoming(d)} w_e * in[src_e, c]
// Thread = (node, class). A wave's 32 lanes share one node -> uniform loop
// bound, wave-uniform csr_src/csr_w loads (broadcast), coalesced 128B gather
// of the feature row (L2-resident: y is 4MB vs 192MB L2). No atomics.
__global__ void prop_gather(const int* __restrict__ row_start, const int* __restrict__ csr_src,
                            const float* __restrict__ csr_w,
                            const float* __restrict__ in, float* __restrict__ out) {
    int gid = blockIdx.x * blockDim.x + threadIdx.x;
    int d = gid >> 6;
    int c = gid & 63;
    int lo = row_start[d], hi = row_start[d + 1];
    float sum = 0.0f;
    for (int i = lo; i < hi; ++i) {
        sum = fmaf(csr_w[i], in[(size_t)csr_src[i] * N_CLASSES + c], sum);
    }
    out[gid] = (1.0f - ALPHA) * in[gid] + ALPHA * sum;
}

// ---------------------------------------------------------------- WMMA GEMM
// out[16384 x 64] = Km[16384 x 16384] @ y[16384 x 64], exact f32 via
// V_WMMA_F32_16X16X4_F32. Wave tile 16x32 (2 accumulators), 8 waves/block.
__launch_bounds__(256)
__global__ void gemm_kernel(const float* __restrict__ Km, const float* __restrict__ y,
                            float* __restrict__ out) {
    const int wv   = threadIdx.x >> 5;
    const int lane = threadIdx.x & 31;
    const int hf   = lane >> 4;        // 0 or 1
    const int r    = lane & 15;
    const int mw   = wv >> 1;          // 0..3
    const int nw   = wv & 1;           // 0..1
    const int m0   = blockIdx.x * 64 + mw * 16;
    const int n0   = nw * 32;

    // A (16x4 f32): lanes 0-15 hold K=0,1 / lanes 16-31 hold K=2,3; M = lane&15
    const float* arow  = Km + (size_t)(m0 + r) * N_NODES + (size_t)(hf * 2);
    // B (4x16 f32): VGPR0 = row K=0 / K=2, VGPR1 = row K=1 / K=3; N = lane&15
    const float* ybase = y + (size_t)(hf * 2) * N_CLASSES + n0 + r;

    v8f acc0 = {};
    v8f acc1 = {};

#pragma unroll 4
    for (int k = 0; k < N_NODES; k += 4) {
        // K is 1 GiB streamed once: keep it out of L2 (y lives there)
        v2f a = __builtin_nontemporal_load((const v2f*)(arow + k));

        const float* yk = ybase + (size_t)k * N_CLASSES;
        v2f b0 = { yk[0],  yk[N_CLASSES] };
        v2f b1 = { yk[16], yk[16 + N_CLASSES] };

        acc0 = __builtin_amdgcn_wmma_f32_16x16x4_f32(
            false, a, false, b0, (short)0, acc0, false, false);
        acc1 = __builtin_amdgcn_wmma_f32_16x16x4_f32(
            false, a, false, b1, (short)0, acc1, false, false);
    }

    // C/D: VGPR j = row (j + 8*hf), col = lane&15
    float* orow = out + (size_t)(m0 + hf * 8) * N_CLASSES + n0 + r;
#pragma unroll
    for (int j = 0; j < 8; ++j) {
        orow[(size_t)j * N_CLASSES + 0]  = acc0[j];
        orow[(size_t)j * N_CLASSES + 16] = acc1[j];
    }
}

// ---------------------------------------------------------------- launcher

extern "C" void kernel_launch(void* const* d_in, const int* in_sizes, int n_in,
                              void* d_out, int out_size, void* d_ws, size_t ws_size,
                              hipStream_t stream) {
    const float* y0  = (const float*)d_in[0];            // [N, C]
    const float* Km  = (const float*)d_in[1];            // [N, N]
    const int*   ei  = (const int*)d_in[2];              // [2, E]
    const int*   src = ei;
    const int*   dst = ei + N_EDGES;
    float*       out = (float*)d_out;

    // workspace layout (all 4-byte elems):
    char* ws = (char*)d_ws;
    int*   cnt       = (int*)ws;                               ws += sizeof(int)   * N_NODES;
    float* dinv      = (float*)ws;                             ws += sizeof(float) * N_NODES;
    int*   row_start = (int*)ws;                               ws += sizeof(int)   * (N_NODES + 1);
    int*   cursor    = (int*)ws;                               ws += sizeof(int)   * N_NODES;
    int*   csr_eid   = (int*)ws;                               ws += sizeof(int)   * N_EDGES;
    int*   csr_src   = (int*)ws;                               ws += sizeof(int)   * N_EDGES;
    float* csr_w     = (float*)ws;                             ws += sizeof(float) * N_EDGES;
    float* bufA      = (float*)ws;                             ws += sizeof(float) * N_NODES * N_CLASSES;
    float* bufB      = (float*)ws;                             ws += sizeof(float) * N_NODES * N_CLASSES;
    float* bufC      = (float*)ws;

    const int NC = N_NODES * N_CLASSES;                        // 1,048,576

    // ---- CSR build (once per call, deterministic after per-node sort)
    zero_u32     <<<N_NODES / 256, 256, 0, stream>>>(cnt, N_NODES);
    count_kernel <<<N_EDGES / 256, 256, 0, stream>>>(dst, cnt);
    dinv_kernel  <<<N_NODES / 256, 256, 0, stream>>>(cnt, dinv);
    scan_kernel  <<<1, 256, 0, stream>>>(cnt, row_start, cursor);
    fill_kernel  <<<N_EDGES / 256, 256, 0, stream>>>(dst, cursor, csr_eid);
    sort_mat_kernel<<<N_NODES / 256, 256, 0, stream>>>(row_start, csr_eid, src, dinv,
                                                       csr_src, csr_w);

    float* bufs[2] = { bufA, bufB };

    // ---- first propagation: y0 -> ping-pong
    const float* cur = y0;
    for (int s = 0; s < KSTEPS; ++s) {
        float* nxt = bufs[s & 1];
        prop_gather<<<NC / 256, 256, 0, stream>>>(row_start, csr_src, csr_w, cur, nxt);
        cur = nxt;
    }

    // ---- dense GEMM: bufC = K @ y
    gemm_kernel<<<N_NODES / 64, 256, 0, stream>>>(Km, cur, bufC);
    cur = bufC;

    // ---- second propagation: final step writes d_out directly
    for (int s = 0; s < KSTEPS; ++s) {
        float* nxt = (s == KSTEPS - 1) ? out : bufs[s & 1];
        prop_gather<<<NC / 256, 256, 0, stream>>>(row_start, csr_src, csr_w, cur, nxt);
        cur = nxt;
    }
}